// GATLayerEdgeSoftmax_44890998177866
// MI455X (gfx1250) — compile-verified
//
#include <hip/hip_runtime.h>

// GAT edge-softmax layer, algebraically factored:
//   u[n] = x[n] @ W_in[:64]          (per node)
//   v[n] = x[n] @ W_in[64:] + b_in   (per node)
//   w[n] = x[n] @ W_out    + b_out   (per node)
//   e_edge = a . leakyrelu(u[src]+v[tgt]) + a_b
//   S[n]   = sum_{tgt==n} exp(e_edge)
//   out[n] = leakyrelu( S[n]/(S[n]+eps) * w[n] )   (softmax collapses per node)
//
// Kernel 1: [N x 64] @ [64 x 192] GEMM via V_WMMA_F32_16X16X4_F32, with the
//           48 KB weight block staged into LDS via global_load_async_to_lds.
// Kernel 2: coalesced edge gather (8 lanes/edge) + exp + L2 atomics.

#define D 64
#define NCOLS 192           // u(64) | v(64) | w(64)
#define EPS 1e-6f
#define SLOPE 0.01f

typedef __attribute__((ext_vector_type(2))) float v2f;
typedef __attribute__((ext_vector_type(8))) float v8f;

#if __has_builtin(__builtin_amdgcn_global_load_async_to_lds_b32) && \
    __has_builtin(__builtin_amdgcn_s_wait_asynccnt)
#define HAVE_ASYNC_LDS 1
typedef __attribute__((address_space(1))) int g_i32;   // global int
typedef __attribute__((address_space(3))) int s_i32;   // LDS int
#else
#define HAVE_ASYNC_LDS 0
#endif

// ---------------------------------------------------------------- node GEMM
__global__ __launch_bounds__(256) void node_gemm_wmma(
    const float* __restrict__ x,      // [n][64]
    const float* __restrict__ W_in,   // [128][64] row-major
    const float* __restrict__ b_in,   // [64]
    const float* __restrict__ W_out,  // [64][64]
    const float* __restrict__ b_out,  // [64]
    float* __restrict__ u,            // [n][64]
    float* __restrict__ v,            // [n][64]
    float* __restrict__ w,            // [n][64]
    int n)
{
  // Weights staged transposed: lds_w[col][k], col in [0,192), k in [0,64).
  __shared__ float lds_w[NCOLS * D];   // 48 KB of the 320 KB WGP LDS
  const int tid = threadIdx.x;
#if HAVE_ASYNC_LDS
  // Async global->LDS copy (ASYNCcnt-tracked); per-lane addresses perform the
  // [k][col] -> [col][k] transpose during the DMA, no VGPR round-trip.
  for (int idx = tid; idx < NCOLS * D; idx += 256) {
    const int col = idx >> 6;
    const int k   = idx & 63;
    const float* gsrc;
    if (col < 64)        gsrc = &W_in[k * D + col];               // W1a
    else if (col < 128)  gsrc = &W_in[(64 + k) * D + (col - 64)]; // W1b
    else                 gsrc = &W_out[k * D + (col - 128)];      // W2
    __builtin_amdgcn_global_load_async_to_lds_b32(
        (g_i32*)gsrc, (s_i32*)&lds_w[col * D + k], 0, 0);
  }
  __builtin_amdgcn_s_wait_asynccnt(0);
#else
  for (int idx = tid; idx < NCOLS * D; idx += 256) {
    const int col = idx >> 6;
    const int k   = idx & 63;
    float val;
    if (col < 64)        val = W_in[k * D + col];
    else if (col < 128)  val = W_in[(64 + k) * D + (col - 64)];
    else                 val = W_out[k * D + (col - 128)];
    lds_w[col * D + k] = val;
  }
#endif
  __syncthreads();

  const int wave = tid >> 5;
  const int lane = tid & 31;
  const int row0 = blockIdx.x * 128 + wave * 16;   // 16 rows per wave
  if (row0 >= n) return;                           // wave-uniform exit
  const bool fulltile = (row0 + 16 <= n);          // wave-uniform

  const int m    = lane & 15;
  const int koff = (lane < 16) ? 0 : 2;   // A frag: lanes<16 hold K=0,1; else K=2,3

  int myrow = row0 + m;
  if (myrow >= n) myrow = n - 1;                   // clamp loads, guard stores
  const float* xrow = x + (size_t)myrow * D;

  // Preload this wave's A tile: 16 K-chunks of (2 x f32) per lane, reused 12x.
  v2f areg[16];
#pragma unroll
  for (int kc = 0; kc < 16; ++kc) {
    const float2 t2 = *(const float2*)(xrow + kc * 4 + koff);
    areg[kc] = (v2f){t2.x, t2.y};
  }

#pragma unroll 1
  for (int ct = 0; ct < 12; ++ct) {                // 12 column tiles of 16
    const int col = ct * 16 + m;
    const int j   = col & 63;
    float bias;
    float* dst;
    if (ct < 4)      { dst = u; bias = 0.f; }
    else if (ct < 8) { dst = v; bias = b_in[j]; }
    else             { dst = w; bias = b_out[j]; }

    // Fold the bias into the C fragment: a lane's 8 elements share one column.
    v8f c = {bias, bias, bias, bias, bias, bias, bias, bias};
    const float* bw = &lds_w[col * D + koff];
#pragma unroll
    for (int kc = 0; kc < 16; ++kc) {              // K = 64 in steps of 4
      const float2 t2 = *(const float2*)(bw + kc * 4);
      const v2f b = (v2f){t2.x, t2.y};
      c = __builtin_amdgcn_wmma_f32_16x16x4_f32(
          /*neg_a=*/false, areg[kc], /*neg_b=*/false, b,
          /*c_mod=*/(short)0, c, /*reuse_a=*/false, /*reuse_b=*/false);
    }
    // Scatter C tile: VGPR r -> row (row0+r) for lanes<16, (row0+8+r) otherwise.
    const int rbase = row0 + ((lane < 16) ? 0 : 8);
    if (fulltile) {
#pragma unroll
      for (int r = 0; r < 8; ++r)
        dst[(size_t)(rbase + r) * D + j] = c[r];
    } else {
#pragma unroll
      for (int r = 0; r < 8; ++r) {
        const int row = rbase + r;
        if (row < n) dst[(size_t)row * D + j] = c[r];
      }
    }
  }
}

// ---------------------------------------------------------------- init S
__global__ void zero_f32(float* __restrict__ p, int n) {
  const int i = blockIdx.x * blockDim.x + threadIdx.x;
  if (i < n) p[i] = 0.f;
}

// ---------------------------------------------------------------- edge pass
// 8 lanes cooperate on one edge: each lane reads 2 float4s (32 B) so the
// octet covers one full 256 B row coalesced; shfl_xor tree reduces the dot.
__global__ __launch_bounds__(256) void edge_softmax_num(
    const int* __restrict__ src, const int* __restrict__ tgt,
    const float* __restrict__ u, const float* __restrict__ v,
    const float* __restrict__ a_w, const float* __restrict__ a_b,
    float* __restrict__ S, int E)
{
  __shared__ float aw_s[D];
  __shared__ float ab_s;
  if (threadIdx.x < D) aw_s[threadIdx.x] = a_w[threadIdx.x];
  if (threadIdx.x == 0) ab_s = a_b[0];
  __syncthreads();

  const int gid  = blockIdx.x * blockDim.x + threadIdx.x;
  const int edge = gid >> 3;
  const int sub  = gid & 7;
  if (edge >= E) return;

  const int s = src[edge];
  const int t = tgt[edge];
  const float4* up = (const float4*)(u + (size_t)s * D) + sub * 2;
  const float4* vp = (const float4*)(v + (size_t)t * D) + sub * 2;  // v has b_in

  float acc = 0.f;
#pragma unroll
  for (int h = 0; h < 2; ++h) {
    const float4 a4 = up[h];
    const float4 b4 = vp[h];
    const int j = sub * 8 + h * 4;
    float z;
    z = a4.x + b4.x; z = (z >= 0.f) ? z : SLOPE * z; acc = fmaf(aw_s[j + 0], z, acc);
    z = a4.y + b4.y; z = (z >= 0.f) ? z : SLOPE * z; acc = fmaf(aw_s[j + 1], z, acc);
    z = a4.z + b4.z; z = (z >= 0.f) ? z : SLOPE * z; acc = fmaf(aw_s[j + 2], z, acc);
    z = a4.w + b4.w; z = (z >= 0.f) ? z : SLOPE * z; acc = fmaf(aw_s[j + 3], z, acc);
  }
  // Reduce across the 8-lane octet.
  acc += __shfl_xor(acc, 4, 8);
  acc += __shfl_xor(acc, 2, 8);
  acc += __shfl_xor(acc, 1, 8);
  if (sub == 0) atomicAdd(&S[t], __expf(acc + ab_s));
}

// ---------------------------------------------------------------- epilogue
__global__ __launch_bounds__(256) void node_epilogue(
    const float* __restrict__ S, const float* __restrict__ w,
    float* __restrict__ out, int total4)   // total4 = n * 16 (float4 count)
{
  const int i = blockIdx.x * blockDim.x + threadIdx.x;
  if (i >= total4) return;
  const float s  = S[i >> 4];
  const float sc = s / (s + EPS);
  const float4 wv = ((const float4*)w)[i];
  float4 o;
  o.x = sc * wv.x; o.x = (o.x >= 0.f) ? o.x : SLOPE * o.x;
  o.y = sc * wv.y; o.y = (o.y >= 0.f) ? o.y : SLOPE * o.y;
  o.z = sc * wv.z; o.z = (o.z >= 0.f) ? o.z : SLOPE * o.z;
  o.w = sc * wv.w; o.w = (o.w >= 0.f) ? o.w : SLOPE * o.w;
  ((float4*)out)[i] = o;
}

// ---------------------------------------------------------------- launcher
extern "C" void kernel_launch(void* const* d_in, const int* in_sizes, int n_in,
                              void* d_out, int out_size, void* d_ws, size_t ws_size,
                              hipStream_t stream) {
  const float* x     = (const float*)d_in[0];
  const int*   src   = (const int*)d_in[1];
  const int*   tgt   = (const int*)d_in[2];
  const float* W_in  = (const float*)d_in[3];
  const float* b_in  = (const float*)d_in[4];
  const float* a_w   = (const float*)d_in[5];
  const float* a_b   = (const float*)d_in[6];
  const float* W_out = (const float*)d_in[7];
  const float* b_out = (const float*)d_in[8];
  float* out = (float*)d_out;

  const int n = in_sizes[0] / D;      // 100000
  const int E = in_sizes[1];          // 1600000

  // Workspace: u | v | w (each n*64 f32) | S (n f32)  ~= 77 MB
  float* u = (float*)d_ws;
  float* v = u + (size_t)n * D;
  float* w = v + (size_t)n * D;
  float* S = w + (size_t)n * D;

  zero_f32<<<(n + 255) / 256, 256, 0, stream>>>(S, n);
  node_gemm_wmma<<<(n + 127) / 128, 256, 0, stream>>>(
      x, W_in, b_in, W_out, b_out, u, v, w, n);
  edge_softmax_num<<<(E + 31) / 32, 256, 0, stream>>>(
      src, tgt, u, v, a_w, a_b, S, E);
  node_epilogue<<<((n * 16) + 255) / 256, 256, 0, stream>>>(S, w, out, n * 16);
}